// SampleCV_7876970021375
// MI455X (gfx1250) — compile-verified
//
#include <hip/hip_runtime.h>

// ---------------------------------------------------------------------------
// Local cost volume (RAFT-style) for MI455X / gfx1250.
//
// Strategy:
//  * Per pixel, bilinear weights are shared across all 81 displacements
//    (integer shifts). Compute integer-grid correlations CI on a 10-row x
//    12-col (2-aligned) window, then OUT = Wy * CI * Wx (separable bilinear).
//  * One wave32 per pixel. Lanes 0..29 each own 4 window columns of one row;
//    fp32 FMA main loop with 8B-aligned float2 global gathers (f2 is fully
//    L2-resident: 63MB << 192MB). f1 pixel vectors staged in LDS, broadcast.
//  * x-direction combine T = CI x Wx done with V_WMMA_F32_16X16X4_F32
//    (exact fp32, 3 K-chunks), D scattered via LDS (documented C/D layout),
//    y-direction combine + stores scalar.
// ---------------------------------------------------------------------------

typedef float v2f __attribute__((ext_vector_type(2)));
typedef float v8f __attribute__((ext_vector_type(8)));

namespace {
constexpr int Nn  = 4;
constexpr int Cc  = 256;
constexpr int Hh  = 96;
constexpr int Ww  = 160;
constexpr int RAD = 4;                         // rx = ry = 4
constexpr int DD  = (2 * RAD + 1) * (2 * RAD + 1);  // 81
constexpr int HW  = Hh * Ww;
constexpr int PIX_PER_BLK = 8;                 // 8 waves -> 8 pixels / block
}

__global__ __launch_bounds__(256)
void cost_volume_kernel(const float* __restrict__ f1,
                        const float* __restrict__ f2,
                        const float* __restrict__ ofs,
                        float* __restrict__ out)
{
    __shared__ float f1s[PIX_PER_BLK][Cc];   // staged f1 pixel vectors
    __shared__ float cib[PIX_PER_BLK][256];  // 16x16 CI grid, reused as T

    const int lane = threadIdx.x;            // 0..31
    const int wv   = threadIdx.y;            // 0..7
    const int t    = wv * 32 + lane;

    // ---- cooperative staging of f1 for the block's 8 pixels ----
    {
        const int p   = t & 7;
        const int c0  = t >> 3;              // 0..31
        const int pix = blockIdx.x * PIX_PER_BLK + p;
        const int px  = pix % Ww;
        const int py  = (pix / Ww) % Hh;
        const int pn  = pix / (Ww * Hh);
        const float* src = f1 + (((size_t)pn * Cc) * Hh + py) * Ww + px;
#pragma unroll
        for (int k = 0; k < 8; ++k) {
            const int c = c0 + 32 * k;
            f1s[p][c] = src[(size_t)c * HW];
        }
    }
    __syncthreads();

    // ---- per-wave pixel ----
    const int p   = wv;
    const int pix = blockIdx.x * PIX_PER_BLK + p;
    const int x   = pix % Ww;
    const int y   = (pix / Ww) % Hh;
    const int n   = pix / (Ww * Hh);

    const float oxv = ofs[(((size_t)n * 2 + 0) * Hh + y) * Ww + x];
    const float oyv = ofs[(((size_t)n * 2 + 1) * Hh + y) * Ww + x];
    const float xcf = (float)x + oxv;
    const float ycf = (float)y + oyv;
    const float fx0 = floorf(xcf);
    const float fy0 = floorf(ycf);
    const float wx  = xcf - fx0;             // shared by all 81 displacements
    const float wyv = ycf - fy0;
    const int   x0  = (int)fx0;
    const int   y0  = (int)fy0;
    const int   gx  = x0 - RAD;              // window origin (cols gx..gx+9)
    const int   gy  = y0 - RAD;              // window rows gy..gy+9
    const int   xw0  = gx & ~1;              // 2-aligned window start
    const int   xoff = gx - xw0;             // 0 or 1

    // zero CI grid (in-wave LDS ordering guarantees visibility)
#pragma unroll
    for (int k = 0; k < 8; ++k) cib[p][lane + 32 * k] = 0.0f;

    // ---- integer-grid correlations: 10 rows x 12 cols, fp32 FMA ----
    if (lane < 30) {
        const int r    = lane / 3;                 // grid row 0..9
        const int colb = (lane - 3 * r) * 4;       // 0,4,8
        const int yi   = gy + r;
        const int yic  = min(max(yi, 0), Hh - 1);
        const int xq   = xw0 + colb;               // even
        const int xa   = min(max(xq, 0),     Ww - 2);  // even -> 8B aligned
        const int xb   = min(max(xq + 2, 0), Ww - 2);
        const float* ba = f2 + (((size_t)n * Cc) * Hh + yic) * Ww + xa;
        const float* bb = f2 + (((size_t)n * Cc) * Hh + yic) * Ww + xb;

        float a0 = 0.f, a1 = 0.f, a2 = 0.f, a3 = 0.f;
        for (int c = 0; c < Cc; c += 4) {
            const float4 fq = *(const float4*)&f1s[p][c];   // LDS broadcast
#pragma unroll
            for (int u = 0; u < 4; ++u) {
                const float  fc = (u == 0) ? fq.x : (u == 1) ? fq.y
                                 : (u == 2) ? fq.z : fq.w;
                const size_t co = (size_t)(c + u) * HW;
                const float2 va = *(const float2*)(ba + co);
                const float2 vb = *(const float2*)(bb + co);
                a0 = fmaf(fc, va.x, a0);
                a1 = fmaf(fc, va.y, a1);
                a2 = fmaf(fc, vb.x, a2);
                a3 = fmaf(fc, vb.y, a3);
            }
        }
        // zero out-of-image samples (reference zero-pads), store CI row chunk
        const bool vy = (yi >= 0) && (yi < Hh);
        float4 o;
        o.x = (vy && (xq + 0) >= 0 && (xq + 0) < Ww) ? a0 : 0.0f;
        o.y = (vy && (xq + 1) >= 0 && (xq + 1) < Ww) ? a1 : 0.0f;
        o.z = (vy && (xq + 2) >= 0 && (xq + 2) < Ww) ? a2 : 0.0f;
        o.w = (vy && (xq + 3) >= 0 && (xq + 3) < Ww) ? a3 : 0.0f;
        *(float4*)&cib[p][r * 16 + colb] = o;    // 16B-aligned LDS store
    }
    // reconverged: EXEC is all-ones again (WMMA requirement)

#if __has_builtin(__builtin_amdgcn_wmma_f32_16x16x4_f32)
    // ---- T = CI x Wx on the matrix unit (exact fp32, 16x16x4, 3 chunks) ----
    {
        const int   half = lane >> 4;
        const int   mm   = lane & 15;       // A: M row / B,D: N col
        const float omwx = 1.0f - wx;
        v8f acc = {0.f, 0.f, 0.f, 0.f, 0.f, 0.f, 0.f, 0.f};
#pragma unroll
        for (int ch = 0; ch < 3; ++ch) {    // cc = 0..11 (Wx rows >=12 are 0)
            const int kb = ch * 4 + 2 * half;
            v2f A, B;
            A.x = cib[p][mm * 16 + kb];
            A.y = cib[p][mm * 16 + kb + 1];
            // Wx[cc][cx] = (1-wx) if cc==cx+xoff, wx if cc==cx+xoff+1
            B.x = (kb     == mm + xoff) ? omwx
                : (kb     == mm + xoff + 1) ? wx : 0.0f;
            B.y = (kb + 1 == mm + xoff) ? omwx
                : (kb + 1 == mm + xoff + 1) ? wx : 0.0f;
            acc = __builtin_amdgcn_wmma_f32_16x16x4_f32(
                      false, A, false, B, (short)0, acc, false, false);
        }
        // scatter D (=T, rows 0..9 valid) to LDS per documented C/D layout:
        // VGPR j holds row j (lanes 0-15) / row j+8 (lanes 16-31), col = N
#pragma unroll
        for (int j = 0; j < 8; ++j)
            cib[p][(j + 8 * half) * 16 + mm] = acc[j];
    }
    // ---- y-direction combine + store ----
    {
        const float omwy = 1.0f - wyv;
#pragma unroll
        for (int i = 0; i < 3; ++i) {
            const int dd = lane + 32 * i;
            if (dd < DD) {
                const int r  = dd / 9;       // dy + 4
                const int cx = dd - 9 * r;   // dx + 4
                const float v = omwy * cib[p][r * 16 + cx]
                              + wyv  * cib[p][(r + 1) * 16 + cx];
                out[(((size_t)n * DD + dd) * Hh + y) * Ww + x] = v;
            }
        }
    }
#else
    // ---- fallback: scalar 4-corner bilinear combine ----
    {
        const float omwx = 1.0f - wx, omwy = 1.0f - wyv;
#pragma unroll
        for (int i = 0; i < 3; ++i) {
            const int dd = lane + 32 * i;
            if (dd < DD) {
                const int r   = dd / 9;
                const int cx  = dd - 9 * r;
                const int col = cx + xoff;
                const float v =
                    omwy * (omwx * cib[p][r * 16 + col]
                          + wx   * cib[p][r * 16 + col + 1])
                  + wyv  * (omwx * cib[p][(r + 1) * 16 + col]
                          + wx   * cib[p][(r + 1) * 16 + col + 1]);
                out[(((size_t)n * DD + dd) * Hh + y) * Ww + x] = v;
            }
        }
    }
#endif
}

extern "C" void kernel_launch(void* const* d_in, const int* in_sizes, int n_in,
                              void* d_out, int out_size, void* d_ws, size_t ws_size,
                              hipStream_t stream) {
    (void)in_sizes; (void)n_in; (void)out_size; (void)d_ws; (void)ws_size;
    const float* f1  = (const float*)d_in[0];
    const float* f2  = (const float*)d_in[1];
    const float* ofs = (const float*)d_in[2];
    float* out = (float*)d_out;

    dim3 block(32, PIX_PER_BLK);                       // 8 wave32 per block
    dim3 grid((Nn * Hh * Ww) / PIX_PER_BLK);           // 7680 blocks
    cost_volume_kernel<<<grid, block, 0, stream>>>(f1, f2, ofs, out);
}